// GraphNetBlock_37632503447812
// MI455X (gfx1250) — compile-verified
//
#include <hip/hip_runtime.h>

typedef __attribute__((ext_vector_type(2))) float v2f;
typedef __attribute__((ext_vector_type(8))) float v8f;

#define LATENT 128
#define ROWS   64      // rows (edges/nodes) per workgroup
#define THREADS 256    // 8 wave32
#define XS_E   388     // LDS stride for edge input (384 + 4 pad -> 4-bank row skew)
#define XS_N   260     // LDS stride for node input (256 + 4 pad)
#define HS     132     // LDS stride for hidden  (128 + 4 pad)
#define EPSV   1e-5f

__device__ __forceinline__ v8f zero8() {
  v8f z;
#pragma unroll
  for (int i = 0; i < 8; ++i) z[i] = 0.0f;
  return z;
}

// ---------------------------------------------------------------------------
// Weight repack: row-major W[K][128]  ->  packed fragment order.
// For k-group sg (8 K-values = two 16x16x4 WMMA steps), lane (n, h) needs
//   W[8sg+2h][n], W[8sg+2h+1][n], W[8sg+4+2h][n], W[8sg+5+2h][n]
// packed contiguously so the inner loop does one coalesced b128 per tile.
// dst float4 index = (sg*128 + n)*2 + h.
// ---------------------------------------------------------------------------
__global__ void repack_kernel(const float* __restrict__ src,
                              float4* __restrict__ dst, int K) {
  const int idx = blockIdx.x * blockDim.x + threadIdx.x;
  const int total = K * LATENT / 4;
  if (idx >= total) return;
  const int h  = idx & 1;
  const int n  = (idx >> 1) & (LATENT - 1);
  const int sg = idx >> 8;
  const int r  = sg * 8 + 2 * h;
  float4 v;
  v.x = src[(r + 0) * LATENT + n];
  v.y = src[(r + 1) * LATENT + n];
  v.z = src[(r + 4) * LATENT + n];
  v.w = src[(r + 5) * LATENT + n];
  dst[idx] = v;
}

// One wave computes a 16x64 output tile as 4x (16x16) WMMA accumulators.
// arow: LDS pointer to this lane's A row (row = rtbase + (lane&15)).
// A frag (16x4 f32): lanes 0-15 -> K = k,k+1 ; lanes 16-31 -> K = k+2,k+3.
// B frags come from the packed weight buffer (one b128 per tile per 8 K).
template <int K>
__device__ __forceinline__ void wmma_gemm(const float* __restrict__ arow,
                                          const float4* __restrict__ Wp,
                                          int colBase, int mn, int half,
                                          v8f acc[4]) {
  const float4* wbase = Wp + (colBase + mn) * 2 + half;
#pragma unroll 2
  for (int k = 0; k < K; k += 8) {
    const int sg = k >> 3;
    const int kr = k + 2 * half;
    v2f a0, a1;
    a0.x = arow[kr];
    a0.y = arow[kr + 1];
    a1.x = arow[kr + 4];
    a1.y = arow[kr + 5];
    float4 w[4];
#pragma unroll
    for (int t = 0; t < 4; ++t) w[t] = wbase[sg * 256 + t * 32];
#pragma unroll
    for (int t = 0; t < 4; ++t) {
      v2f b;
      b.x = w[t].x;
      b.y = w[t].y;
      acc[t] = __builtin_amdgcn_wmma_f32_16x16x4_f32(
          false, a0, false, b, (short)0, acc[t], false, false);
    }
#pragma unroll
    for (int t = 0; t < 4; ++t) {
      v2f b;
      b.x = w[t].z;
      b.y = w[t].w;
      acc[t] = __builtin_amdgcn_wmma_f32_16x16x4_f32(
          false, a1, false, b, (short)0, acc[t], false, false);
    }
  }
}

// C frag: lane&15 = N col; element i -> row rtbase + i + 8*half.
__device__ __forceinline__ void store_act(float* __restrict__ h,
                                          const float* __restrict__ bias,
                                          int colBase, int mn, int half,
                                          int rtbase, v8f acc[4], bool relu) {
#pragma unroll
  for (int t = 0; t < 4; ++t) {
    const int col = colBase + t * 16 + mn;
    const float bv = bias[col];
#pragma unroll
    for (int i = 0; i < 8; ++i) {
      float v = acc[t][i] + bv;
      if (relu) v = fmaxf(v, 0.0f);
      h[(rtbase + i + 8 * half) * HS + col] = v;
    }
  }
}

__global__ void zero_kernel(float4* __restrict__ p, long long n4) {
  long long i = (long long)blockIdx.x * blockDim.x + threadIdx.x;
  if (i < n4) p[i] = make_float4(0.f, 0.f, 0.f, 0.f);
}

__global__ __launch_bounds__(THREADS)
void edge_mlp_kernel(const float* __restrict__ nodef,
                     const float* __restrict__ edgef,
                     const int* __restrict__ senders,
                     const int* __restrict__ receivers,
                     const float4* __restrict__ W0, const float* __restrict__ b0,
                     const float4* __restrict__ W1, const float* __restrict__ b1,
                     const float4* __restrict__ W2, const float* __restrict__ b2,
                     const float* __restrict__ gam, const float* __restrict__ bet,
                     float* __restrict__ out_edge, float* __restrict__ seg_sum,
                     int E) {
  extern __shared__ float smem[];
  float* xs   = smem;                   // ROWS * XS_E (also reused as h2, stride HS)
  float* hbuf = xs + ROWS * XS_E;       // ROWS * HS
  float* ps   = hbuf + ROWS * HS;       // ROWS * 4
  float* pq   = ps + ROWS * 4;          // ROWS * 4
  int*   sidx = (int*)(pq + ROWS * 4);  // ROWS
  int*   ridx = sidx + ROWS;            // ROWS

  const int tid     = threadIdx.x;
  const int lane    = tid & 31;
  const int wave    = tid >> 5;
  const int mn      = lane & 15;
  const int half    = lane >> 4;
  const int rtbase  = (wave >> 1) * 16;
  const int colBase = (wave & 1) * 64;
  const int e0      = blockIdx.x * ROWS;

  if (tid < ROWS) {
    const int e = e0 + tid;
    sidx[tid] = (e < E) ? senders[e] : 0;
    ridx[tid] = (e < E) ? receivers[e] : 0;
  }
  __syncthreads();

  // Stage gathered [sender | receiver | edge] rows: ROWS x 384, float4 chunks.
  for (int idx = tid; idx < ROWS * 96; idx += THREADS) {
    const int row = idx / 96;
    const int q   = idx % 96;
    const int e   = e0 + row;
    float4 v = make_float4(0.f, 0.f, 0.f, 0.f);
    if (e < E) {
      if (q < 32)      v = *(const float4*)(nodef + sidx[row] * LATENT + q * 4);
      else if (q < 64) v = *(const float4*)(nodef + ridx[row] * LATENT + (q - 32) * 4);
      else             v = *(const float4*)(edgef + e * LATENT + (q - 64) * 4);
    }
    *(float4*)(xs + row * XS_E + q * 4) = v;
  }
  __syncthreads();

  v8f acc[4];
#pragma unroll
  for (int t = 0; t < 4; ++t) acc[t] = zero8();
  wmma_gemm<384>(xs + (rtbase + mn) * XS_E, W0, colBase, mn, half, acc);
  store_act(hbuf, b0, colBase, mn, half, rtbase, acc, true);
  __syncthreads();

#pragma unroll
  for (int t = 0; t < 4; ++t) acc[t] = zero8();
  wmma_gemm<128>(hbuf + (rtbase + mn) * HS, W1, colBase, mn, half, acc);
  float* h2 = xs;  // reuse input buffer (stride HS)
  store_act(h2, b1, colBase, mn, half, rtbase, acc, true);
  __syncthreads();

#pragma unroll
  for (int t = 0; t < 4; ++t) acc[t] = zero8();
  wmma_gemm<128>(h2 + (rtbase + mn) * HS, W2, colBase, mn, half, acc);
  store_act(hbuf, b2, colBase, mn, half, rtbase, acc, false);
  __syncthreads();

  // LayerNorm partials: 4 threads per row, 32 cols each.
  {
    const int row = tid >> 2;
    const int seg = tid & 3;
    const float* hr = hbuf + row * HS + seg * 32;
    float s = 0.f, q2 = 0.f;
#pragma unroll
    for (int j = 0; j < 8; ++j) {
      float4 v = *(const float4*)(hr + j * 4);
      s  += v.x + v.y + v.z + v.w;
      q2 += v.x * v.x + v.y * v.y + v.z * v.z + v.w * v.w;
    }
    ps[row * 4 + seg] = s;
    pq[row * 4 + seg] = q2;
  }
  __syncthreads();
  {
    const int row = tid >> 2;
    const int seg = tid & 3;
    const int e   = e0 + row;
    if (e < E) {
      const float s   = ps[row*4+0] + ps[row*4+1] + ps[row*4+2] + ps[row*4+3];
      const float q2  = pq[row*4+0] + pq[row*4+1] + pq[row*4+2] + pq[row*4+3];
      const float mu  = s * (1.0f / LATENT);
      const float var = q2 * (1.0f / LATENT) - mu * mu;
      const float rstd = rsqrtf(var + EPSV);
      const int rcv = ridx[row];
      const float* hr = hbuf + row * HS + seg * 32;
#pragma unroll
      for (int j = 0; j < 8; ++j) {
        const int c = seg * 32 + j * 4;
        float4 h  = *(const float4*)(hr + j * 4);
        float4 gg = *(const float4*)(gam + c);
        float4 bb = *(const float4*)(bet + c);
        float4 y;
        y.x = gg.x * (h.x - mu) * rstd + bb.x;
        y.y = gg.y * (h.y - mu) * rstd + bb.y;
        y.z = gg.z * (h.z - mu) * rstd + bb.z;
        y.w = gg.w * (h.w - mu) * rstd + bb.w;
        float4 ef = *(const float4*)(edgef + e * LATENT + c);
        float4 o  = make_float4(y.x + ef.x, y.y + ef.y, y.z + ef.z, y.w + ef.w);
        *(float4*)(out_edge + e * LATENT + c) = o;
        float* sp = seg_sum + rcv * LATENT + c;
        atomicAdd(sp + 0, y.x);
        atomicAdd(sp + 1, y.y);
        atomicAdd(sp + 2, y.z);
        atomicAdd(sp + 3, y.w);
      }
    }
  }
}

__global__ __launch_bounds__(THREADS)
void node_mlp_kernel(const float* __restrict__ nodef,
                     const float* __restrict__ seg_sum,
                     const float4* __restrict__ W0, const float* __restrict__ b0,
                     const float4* __restrict__ W1, const float* __restrict__ b1,
                     const float4* __restrict__ W2, const float* __restrict__ b2,
                     const float* __restrict__ gam, const float* __restrict__ bet,
                     float* __restrict__ out_node, int N) {
  extern __shared__ float smem[];
  float* xs   = smem;              // ROWS * XS_N (reused as h2, stride HS)
  float* hbuf = xs + ROWS * XS_N;  // ROWS * HS
  float* ps   = hbuf + ROWS * HS;  // ROWS * 4
  float* pq   = ps + ROWS * 4;     // ROWS * 4

  const int tid     = threadIdx.x;
  const int lane    = tid & 31;
  const int wave    = tid >> 5;
  const int mn      = lane & 15;
  const int half    = lane >> 4;
  const int rtbase  = (wave >> 1) * 16;
  const int colBase = (wave & 1) * 64;
  const int n0      = blockIdx.x * ROWS;

  // Stage [node_features | seg_sum]: ROWS x 256.
  for (int idx = tid; idx < ROWS * 64; idx += THREADS) {
    const int row = idx >> 6;
    const int q   = idx & 63;
    const int n   = n0 + row;
    float4 v = make_float4(0.f, 0.f, 0.f, 0.f);
    if (n < N) {
      if (q < 32) v = *(const float4*)(nodef + n * LATENT + q * 4);
      else        v = *(const float4*)(seg_sum + n * LATENT + (q - 32) * 4);
    }
    *(float4*)(xs + row * XS_N + q * 4) = v;
  }
  __syncthreads();

  v8f acc[4];
#pragma unroll
  for (int t = 0; t < 4; ++t) acc[t] = zero8();
  wmma_gemm<256>(xs + (rtbase + mn) * XS_N, W0, colBase, mn, half, acc);
  store_act(hbuf, b0, colBase, mn, half, rtbase, acc, true);
  __syncthreads();

#pragma unroll
  for (int t = 0; t < 4; ++t) acc[t] = zero8();
  wmma_gemm<128>(hbuf + (rtbase + mn) * HS, W1, colBase, mn, half, acc);
  float* h2 = xs;
  store_act(h2, b1, colBase, mn, half, rtbase, acc, true);
  __syncthreads();

#pragma unroll
  for (int t = 0; t < 4; ++t) acc[t] = zero8();
  wmma_gemm<128>(h2 + (rtbase + mn) * HS, W2, colBase, mn, half, acc);
  store_act(hbuf, b2, colBase, mn, half, rtbase, acc, false);
  __syncthreads();

  {
    const int row = tid >> 2;
    const int seg = tid & 3;
    const float* hr = hbuf + row * HS + seg * 32;
    float s = 0.f, q2 = 0.f;
#pragma unroll
    for (int j = 0; j < 8; ++j) {
      float4 v = *(const float4*)(hr + j * 4);
      s  += v.x + v.y + v.z + v.w;
      q2 += v.x * v.x + v.y * v.y + v.z * v.z + v.w * v.w;
    }
    ps[row * 4 + seg] = s;
    pq[row * 4 + seg] = q2;
  }
  __syncthreads();
  {
    const int row = tid >> 2;
    const int seg = tid & 3;
    const int n   = n0 + row;
    if (n < N) {
      const float s   = ps[row*4+0] + ps[row*4+1] + ps[row*4+2] + ps[row*4+3];
      const float q2  = pq[row*4+0] + pq[row*4+1] + pq[row*4+2] + pq[row*4+3];
      const float mu  = s * (1.0f / LATENT);
      const float var = q2 * (1.0f / LATENT) - mu * mu;
      const float rstd = rsqrtf(var + EPSV);
      const float* hr = hbuf + row * HS + seg * 32;
#pragma unroll
      for (int j = 0; j < 8; ++j) {
        const int c = seg * 32 + j * 4;
        float4 h  = *(const float4*)(hr + j * 4);
        float4 gg = *(const float4*)(gam + c);
        float4 bb = *(const float4*)(bet + c);
        float4 nf = *(const float4*)(nodef + n * LATENT + c);
        float4 o;
        o.x = gg.x * (h.x - mu) * rstd + bb.x + nf.x;
        o.y = gg.y * (h.y - mu) * rstd + bb.y + nf.y;
        o.z = gg.z * (h.z - mu) * rstd + bb.z + nf.z;
        o.w = gg.w * (h.w - mu) * rstd + bb.w + nf.w;
        *(float4*)(out_node + n * LATENT + c) = o;
      }
    }
  }
}

extern "C" void kernel_launch(void* const* d_in, const int* in_sizes, int n_in,
                              void* d_out, int out_size, void* d_ws, size_t ws_size,
                              hipStream_t stream) {
  const float* nodef    = (const float*)d_in[0];
  const float* edgef    = (const float*)d_in[1];
  const int*   senders  = (const int*)d_in[2];
  const int*   receivers= (const int*)d_in[3];
  const float* eW0 = (const float*)d_in[4];
  const float* eb0 = (const float*)d_in[5];
  const float* eW1 = (const float*)d_in[6];
  const float* eb1 = (const float*)d_in[7];
  const float* eW2 = (const float*)d_in[8];
  const float* eb2 = (const float*)d_in[9];
  const float* eg  = (const float*)d_in[10];
  const float* ebt = (const float*)d_in[11];
  const float* nW0 = (const float*)d_in[12];
  const float* nb0 = (const float*)d_in[13];
  const float* nW1 = (const float*)d_in[14];
  const float* nb1 = (const float*)d_in[15];
  const float* nW2 = (const float*)d_in[16];
  const float* nb2 = (const float*)d_in[17];
  const float* ng  = (const float*)d_in[18];
  const float* nbt = (const float*)d_in[19];

  const int E = in_sizes[2];
  const int N = in_sizes[0] / LATENT;

  float* out_node = (float*)d_out;
  float* out_edge = out_node + (long long)N * LATENT;

  // Workspace layout: [seg_sum: N*128] [packed weights: 1152*128]
  float* seg_sum = (float*)d_ws;
  float* pw      = seg_sum + (long long)N * LATENT;
  float4* peW0 = (float4*)pw;                       // 384*128 floats
  float4* peW1 = peW0 + 384 * LATENT / 4;           // 128*128
  float4* peW2 = peW1 + 128 * LATENT / 4;           // 128*128
  float4* pnW0 = peW2 + 128 * LATENT / 4;           // 256*128
  float4* pnW1 = pnW0 + 256 * LATENT / 4;           // 128*128
  float4* pnW2 = pnW1 + 128 * LATENT / 4;           // 128*128

  // 0) repack weights into WMMA B-fragment order (tiny; L2-resident after)
  repack_kernel<<<(384 * LATENT / 4 + 255) / 256, 256, 0, stream>>>(eW0, peW0, 384);
  repack_kernel<<<(128 * LATENT / 4 + 255) / 256, 256, 0, stream>>>(eW1, peW1, 128);
  repack_kernel<<<(128 * LATENT / 4 + 255) / 256, 256, 0, stream>>>(eW2, peW2, 128);
  repack_kernel<<<(256 * LATENT / 4 + 255) / 256, 256, 0, stream>>>(nW0, pnW0, 256);
  repack_kernel<<<(128 * LATENT / 4 + 255) / 256, 256, 0, stream>>>(nW1, pnW1, 128);
  repack_kernel<<<(128 * LATENT / 4 + 255) / 256, 256, 0, stream>>>(nW2, pnW2, 128);

  // 1) zero the scatter accumulator
  const long long n4 = (long long)N * LATENT / 4;
  zero_kernel<<<(int)((n4 + 255) / 256), 256, 0, stream>>>((float4*)seg_sum, n4);

  // 2) edge MLP + residual + scatter-add
  const size_t edge_smem =
      (size_t)(ROWS * XS_E + ROWS * HS + ROWS * 8) * sizeof(float) +
      (size_t)(ROWS * 2) * sizeof(int);
  edge_mlp_kernel<<<(E + ROWS - 1) / ROWS, THREADS, edge_smem, stream>>>(
      nodef, edgef, senders, receivers, peW0, eb0, peW1, eb1, peW2, eb2, eg, ebt,
      out_edge, seg_sum, E);

  // 3) node MLP + residual
  const size_t node_smem =
      (size_t)(ROWS * XS_N + ROWS * HS + ROWS * 8) * sizeof(float);
  node_mlp_kernel<<<(N + ROWS - 1) / ROWS, THREADS, node_smem, stream>>>(
      nodef, seg_sum, pnW0, nb0, pnW1, nb1, pnW2, nb2, ng, nbt, out_node, N);
}